// iSQRTCOVPool_27504970564007
// MI455X (gfx1250) — compile-verified
//
#include <hip/hip_runtime.h>
#include <hip/hip_bf16.h>
#include <math.h>

typedef __attribute__((ext_vector_type(2))) float v2f;
typedef __attribute__((ext_vector_type(8))) float v8f;

#define C_DIM 256
#define N_PIX 196
#define MAT_ELEMS (C_DIM * C_DIM)            // 65536
#define BATCH 64
#define OUT_PER_B 32896                      // 256*257/2
#define LDS_STRIDE 34                        // even (8B-aligned v2f) + bank spread

// ---------------------------------------------------------------------------
// mean over the n=196 pixels, one wave (32 lanes) per row; 8 rows per block
// ---------------------------------------------------------------------------
__global__ __launch_bounds__(256) void mean_kernel(const float* __restrict__ x,
                                                   float* __restrict__ mean) {
    int row = blockIdx.x * 8 + (threadIdx.x >> 5);   // 0 .. 64*256-1
    int lane = threadIdx.x & 31;
    const float* xr = x + (size_t)row * N_PIX;
    float s = 0.0f;
    for (int i = lane; i < N_PIX; i += 32) s += xr[i];
    #pragma unroll
    for (int o = 16; o > 0; o >>= 1) s += __shfl_down(s, o, 32);
    if (lane == 0) mean[row] = s * (1.0f / (float)N_PIX);
}

// ---------------------------------------------------------------------------
// sigma = ((x - mu)/n) @ x^T    [B, 256, 256], K = 196 padded to 224
// WG tile 128x128, 8 waves = 4(m) x 2(n), wave tile 32x64 (2x4 f32 WMMA accs)
// ---------------------------------------------------------------------------
__global__ __launch_bounds__(256) void sigma_kernel(float* __restrict__ D,
                                                    const float* __restrict__ x,
                                                    const float* __restrict__ mean) {
    __shared__ float As[128 * LDS_STRIDE];
    __shared__ float Bt[128 * LDS_STRIDE];
    const int tid  = threadIdx.x;
    const int lane = tid & 31;
    const int wave = tid >> 5;
    const int b    = blockIdx.y;
    const int tileid   = blockIdx.x;          // 0..3
    const int row_base = (tileid >> 1) * 128;
    const int col_base = (tileid & 1) * 128;
    const float* xb = x + (size_t)b * C_DIM * N_PIX;
    const float* mb = mean + (size_t)b * C_DIM;
    float* Db = D + (size_t)b * MAT_ELEMS;

    const int wrow = (wave >> 1) * 32;        // 0..96
    const int wcol = (wave & 1) * 64;         // 0 or 64
    const int lr = lane & 15;
    const int kh = (lane >> 4) * 2;
    const float inv_n = 1.0f / (float)N_PIX;

    const int aR0 = (wrow +      lr) * LDS_STRIDE;
    const int aR1 = (wrow + 16 + lr) * LDS_STRIDE;
    const int bR0 = (wcol +      lr) * LDS_STRIDE;
    const int bR1 = (wcol + 16 + lr) * LDS_STRIDE;
    const int bR2 = (wcol + 32 + lr) * LDS_STRIDE;
    const int bR3 = (wcol + 48 + lr) * LDS_STRIDE;

    v8f acc00 = {}, acc01 = {}, acc02 = {}, acc03 = {};
    v8f acc10 = {}, acc11 = {}, acc12 = {}, acc13 = {};

    for (int kk = 0; kk < 224; kk += 32) {
        __syncthreads();
        // A tile: centered/scaled rows of x  (128 x 32)
        #pragma unroll
        for (int r = 0; r < 16; ++r) {
            int e = r * 256 + tid;
            int mm = e >> 5, k = e & 31;
            int kg = kk + k;
            int row = row_base + mm;
            float v = 0.0f;
            if (kg < N_PIX) v = (xb[(size_t)row * N_PIX + kg] - mb[row]) * inv_n;
            As[mm * LDS_STRIDE + k] = v;
        }
        // B^T tile: raw rows of x (x is already [N][K] layout)  (128 x 32)
        #pragma unroll
        for (int r = 0; r < 16; ++r) {
            int e = r * 256 + tid;
            int nn = e >> 5, k = e & 31;
            int kg = kk + k;
            int col = col_base + nn;
            float v = 0.0f;
            if (kg < N_PIX) v = xb[(size_t)col * N_PIX + kg];
            Bt[nn * LDS_STRIDE + k] = v;
        }
        __syncthreads();
        #pragma unroll
        for (int ks = 0; ks < 8; ++ks) {
            int kf = ks * 4 + kh;
            v2f a0 = *(const v2f*)&As[aR0 + kf];
            v2f a1 = *(const v2f*)&As[aR1 + kf];
            v2f b0 = *(const v2f*)&Bt[bR0 + kf];
            v2f b1 = *(const v2f*)&Bt[bR1 + kf];
            v2f b2 = *(const v2f*)&Bt[bR2 + kf];
            v2f b3 = *(const v2f*)&Bt[bR3 + kf];
            acc00 = __builtin_amdgcn_wmma_f32_16x16x4_f32(false, a0, false, b0, (short)0, acc00, false, false);
            acc01 = __builtin_amdgcn_wmma_f32_16x16x4_f32(false, a0, false, b1, (short)0, acc01, false, false);
            acc02 = __builtin_amdgcn_wmma_f32_16x16x4_f32(false, a0, false, b2, (short)0, acc02, false, false);
            acc03 = __builtin_amdgcn_wmma_f32_16x16x4_f32(false, a0, false, b3, (short)0, acc03, false, false);
            acc10 = __builtin_amdgcn_wmma_f32_16x16x4_f32(false, a1, false, b0, (short)0, acc10, false, false);
            acc11 = __builtin_amdgcn_wmma_f32_16x16x4_f32(false, a1, false, b1, (short)0, acc11, false, false);
            acc12 = __builtin_amdgcn_wmma_f32_16x16x4_f32(false, a1, false, b2, (short)0, acc12, false, false);
            acc13 = __builtin_amdgcn_wmma_f32_16x16x4_f32(false, a1, false, b3, (short)0, acc13, false, false);
        }
    }
    const int mrow = row_base + wrow + (lane >> 4) * 8;
    const int ncol = col_base + wcol + lr;
    #pragma unroll
    for (int v = 0; v < 8; ++v) {
        Db[(size_t)(mrow + v) * C_DIM + ncol]           = acc00[v];
        Db[(size_t)(mrow + v) * C_DIM + ncol + 16]      = acc01[v];
        Db[(size_t)(mrow + v) * C_DIM + ncol + 32]      = acc02[v];
        Db[(size_t)(mrow + v) * C_DIM + ncol + 48]      = acc03[v];
        Db[(size_t)(mrow + 16 + v) * C_DIM + ncol]      = acc10[v];
        Db[(size_t)(mrow + 16 + v) * C_DIM + ncol + 16] = acc11[v];
        Db[(size_t)(mrow + 16 + v) * C_DIM + ncol + 32] = acc12[v];
        Db[(size_t)(mrow + 16 + v) * C_DIM + ncol + 48] = acc13[v];
    }
}

// ---------------------------------------------------------------------------
// trace -> scal[b] = 1/tr, scal[64+b] = sqrt(tr)
// ---------------------------------------------------------------------------
__global__ __launch_bounds__(256) void trace_kernel(const float* __restrict__ S,
                                                    float* __restrict__ scal) {
    __shared__ float red[256];
    int b = blockIdx.x, tid = threadIdx.x;
    red[tid] = S[(size_t)b * MAT_ELEMS + (size_t)tid * 257];
    __syncthreads();
    for (int s = 128; s > 0; s >>= 1) {
        if (tid < s) red[tid] += red[tid + s];
        __syncthreads();
    }
    if (tid == 0) {
        float tr = red[0];
        scal[b] = 1.0f / tr;
        scal[64 + b] = sqrtf(tr);
    }
}

// ---------------------------------------------------------------------------
// b0 = 0.5 * (3I - sigma * (1/tr))
// ---------------------------------------------------------------------------
__global__ __launch_bounds__(256) void b0_kernel(float* __restrict__ Bm,
                                                 const float* __restrict__ S,
                                                 const float* __restrict__ scal) {
    int col = threadIdx.x, row = blockIdx.x, b = blockIdx.y;
    size_t idx = (size_t)b * MAT_ELEMS + (size_t)row * C_DIM + col;
    float rtr = scal[b];
    Bm[idx] = 0.5f * ((row == col ? 3.0f : 0.0f) - S[idx] * rtr);
}

// ---------------------------------------------------------------------------
// Batched 256x256x256 GEMM  D = A @ B  with fused epilogue:
//   op 0: D = acc
//   op 1: D = 0.5*(3I - acc)            (Newton-Schulz "b" step)
//   op 2: D = acc * scal[b]             (per-batch scalar)
//   op 3: out = triu(acc * scal[b])     (final: scale by sqrt(tr) + pack triu)
// WG tile 128x128, 8 waves = 4(m) x 2(n), wave tile 32x64 (2x4 f32 WMMA accs)
// ---------------------------------------------------------------------------
__global__ __launch_bounds__(256) void ns_gemm_kernel(float* __restrict__ D,
                                                      const float* __restrict__ A,
                                                      const float* __restrict__ B,
                                                      const float* __restrict__ scal,
                                                      int op) {
    __shared__ float As[128 * LDS_STRIDE];
    __shared__ float Bt[128 * LDS_STRIDE];
    const int tid  = threadIdx.x;
    const int lane = tid & 31;
    const int wave = tid >> 5;
    const int b    = blockIdx.y;
    const int tileid   = blockIdx.x;          // 0..3
    const int row_base = (tileid >> 1) * 128;
    const int col_base = (tileid & 1) * 128;
    const float* Ab = A + (size_t)b * MAT_ELEMS;
    const float* Bb = B + (size_t)b * MAT_ELEMS;
    float* Dmat = D + (size_t)b * MAT_ELEMS;   // ops 0..2
    float* Dout = D + (size_t)b * OUT_PER_B;   // op 3 (packed triu)

    const int wrow = (wave >> 1) * 32;
    const int wcol = (wave & 1) * 64;
    const int lr = lane & 15;
    const int kh = (lane >> 4) * 2;

    const int aR0 = (wrow +      lr) * LDS_STRIDE;
    const int aR1 = (wrow + 16 + lr) * LDS_STRIDE;
    const int bR0 = (wcol +      lr) * LDS_STRIDE;
    const int bR1 = (wcol + 16 + lr) * LDS_STRIDE;
    const int bR2 = (wcol + 32 + lr) * LDS_STRIDE;
    const int bR3 = (wcol + 48 + lr) * LDS_STRIDE;

    v8f acc00 = {}, acc01 = {}, acc02 = {}, acc03 = {};
    v8f acc10 = {}, acc11 = {}, acc12 = {}, acc13 = {};

    for (int kk = 0; kk < 256; kk += 32) {
        __syncthreads();
        // stage A 128x32 (row-major, stride 34: 8B aligned, conflict-free)
        #pragma unroll
        for (int r = 0; r < 16; ++r) {
            int e = r * 256 + tid;
            int mm = e >> 5, k = e & 31;
            As[mm * LDS_STRIDE + k] = Ab[(size_t)(row_base + mm) * C_DIM + kk + k];
        }
        // stage B^T 128x32 (transpose on the fly; coalesced global reads,
        // conflict-free column scatter with stride 34)
        #pragma unroll
        for (int r = 0; r < 16; ++r) {
            int e = r * 256 + tid;
            int k = e >> 7, nn = e & 127;
            Bt[nn * LDS_STRIDE + k] = Bb[(size_t)(kk + k) * C_DIM + col_base + nn];
        }
        __syncthreads();
        #pragma unroll
        for (int ks = 0; ks < 8; ++ks) {
            int kf = ks * 4 + kh;
            v2f a0 = *(const v2f*)&As[aR0 + kf];
            v2f a1 = *(const v2f*)&As[aR1 + kf];
            v2f b0 = *(const v2f*)&Bt[bR0 + kf];
            v2f b1 = *(const v2f*)&Bt[bR1 + kf];
            v2f b2 = *(const v2f*)&Bt[bR2 + kf];
            v2f b3 = *(const v2f*)&Bt[bR3 + kf];
            acc00 = __builtin_amdgcn_wmma_f32_16x16x4_f32(false, a0, false, b0, (short)0, acc00, false, false);
            acc01 = __builtin_amdgcn_wmma_f32_16x16x4_f32(false, a0, false, b1, (short)0, acc01, false, false);
            acc02 = __builtin_amdgcn_wmma_f32_16x16x4_f32(false, a0, false, b2, (short)0, acc02, false, false);
            acc03 = __builtin_amdgcn_wmma_f32_16x16x4_f32(false, a0, false, b3, (short)0, acc03, false, false);
            acc10 = __builtin_amdgcn_wmma_f32_16x16x4_f32(false, a1, false, b0, (short)0, acc10, false, false);
            acc11 = __builtin_amdgcn_wmma_f32_16x16x4_f32(false, a1, false, b1, (short)0, acc11, false, false);
            acc12 = __builtin_amdgcn_wmma_f32_16x16x4_f32(false, a1, false, b2, (short)0, acc12, false, false);
            acc13 = __builtin_amdgcn_wmma_f32_16x16x4_f32(false, a1, false, b3, (short)0, acc13, false, false);
        }
    }

    float s = 1.0f;
    if (op >= 2) s = scal[b];
    const int mrow = row_base + wrow + (lane >> 4) * 8;
    const int ncol = col_base + wcol + lr;

    auto emit = [&](float val, int m, int n) {
        if (op == 3) {
            if (n >= m) {
                int start = m * C_DIM - (m * (m - 1)) / 2;
                Dout[start + (n - m)] = val * s;
            }
        } else {
            if (op == 1)      val = (m == n ? 1.5f : 0.0f) - 0.5f * val;
            else if (op == 2) val *= s;
            Dmat[(size_t)m * C_DIM + n] = val;
        }
    };
    #pragma unroll
    for (int v = 0; v < 8; ++v) {
        emit(acc00[v], mrow + v,      ncol);
        emit(acc01[v], mrow + v,      ncol + 16);
        emit(acc02[v], mrow + v,      ncol + 32);
        emit(acc03[v], mrow + v,      ncol + 48);
        emit(acc10[v], mrow + 16 + v, ncol);
        emit(acc11[v], mrow + 16 + v, ncol + 16);
        emit(acc12[v], mrow + 16 + v, ncol + 32);
        emit(acc13[v], mrow + 16 + v, ncol + 48);
    }
}

// ---------------------------------------------------------------------------
extern "C" void kernel_launch(void* const* d_in, const int* in_sizes, int n_in,
                              void* d_out, int out_size, void* d_ws, size_t ws_size,
                              hipStream_t stream) {
    const float* x = (const float*)d_in[0];
    float* out = (float*)d_out;
    float* ws = (float*)d_ws;

    float* mean = ws;                      // 64*256
    float* scal = ws + 16384;              // 128 : [0..63]=1/tr, [64..127]=sqrt(tr)
    const size_t MS = (size_t)BATCH * MAT_ELEMS;   // 4,194,304 floats per matrix
    float* M0 = ws + 16512;
    float* M1 = M0 + MS;
    float* M2 = M1 + MS;
    float* M3 = M2 + MS;
    float* M4 = M3 + MS;

    dim3 gGemm(4, BATCH);
    dim3 gRow(C_DIM, BATCH);

    mean_kernel<<<2048, 256, 0, stream>>>(x, mean);
    sigma_kernel<<<gGemm, 256, 0, stream>>>(M0, x, mean);          // S = sigma
    trace_kernel<<<BATCH, 256, 0, stream>>>(M0, scal);
    b0_kernel<<<gRow, 256, 0, stream>>>(M1, M0, scal);             // B0 = 0.5(3I - S/tr)

    // y1 = a @ b0 = (S @ B0) * (1/tr) ; z1 = B0 (alias M1)
    ns_gemm_kernel<<<gGemm, 256, 0, stream>>>(M2, M0, M1, scal, 2);    // Y = M2

    // iter 1: B = 0.5(3I - Z@Y); Y = Y@B; Z = B@Z
    ns_gemm_kernel<<<gGemm, 256, 0, stream>>>(M3, M1, M2, nullptr, 1); // B = M3
    ns_gemm_kernel<<<gGemm, 256, 0, stream>>>(M4, M2, M3, nullptr, 0); // Y = M4
    ns_gemm_kernel<<<gGemm, 256, 0, stream>>>(M0, M3, M1, nullptr, 0); // Z = M0
    // iter 2
    ns_gemm_kernel<<<gGemm, 256, 0, stream>>>(M1, M0, M4, nullptr, 1); // B = M1
    ns_gemm_kernel<<<gGemm, 256, 0, stream>>>(M2, M4, M1, nullptr, 0); // Y = M2
    ns_gemm_kernel<<<gGemm, 256, 0, stream>>>(M3, M1, M0, nullptr, 0); // Z = M3
    // iter 3
    ns_gemm_kernel<<<gGemm, 256, 0, stream>>>(M4, M3, M2, nullptr, 1); // B = M4
    ns_gemm_kernel<<<gGemm, 256, 0, stream>>>(M0, M2, M4, nullptr, 0); // Y = M0
    ns_gemm_kernel<<<gGemm, 256, 0, stream>>>(M1, M4, M3, nullptr, 0); // Z = M1

    // final: bf = 0.5(3I - Z@Y); out = triu((Y @ bf) * sqrt(tr))
    ns_gemm_kernel<<<gGemm, 256, 0, stream>>>(M2, M1, M0, nullptr, 1);     // bf = M2
    ns_gemm_kernel<<<gGemm, 256, 0, stream>>>(out, M0, M2, scal + 64, 3);  // fused
}